// KalmanFilter_72748156059830
// MI455X (gfx1250) — compile-verified
//
#include <hip/hip_runtime.h>
#include <hip/hip_bf16.h>
#include <math.h>

// Problem constants (from reference)
#define BSZ   128
#define TLEN  100
#define DZ    64
#define DA    32
#define DU    8
#define KMIX  8
#define DOBS  16
#define HH    128
#define DIN   (DA + DOBS + DU)   // 56
#define JIT   1.0e-4f
#define NTHR  256
#define NWAVE 8

// Padded LDS leading dimensions: bank = (row*65 + col) % 64 = (row+col) % 64
// -> conflict-free for both row-strided (transposed) and column-strided reads.
#define L64   65   // stride for 64-column LDS matrices
#define L32   33   // stride for 32-column LDS matrices

typedef float v2f __attribute__((ext_vector_type(2)));
typedef float v8f __attribute__((ext_vector_type(8)));

struct SMem {
  float P [DZ*L64];   // carried covariance (P0 / P_pred)
  float A [DZ*L64];   // mixed A
  float T1[DZ*L64];   // temp (IKC, KgRKgT)
  float T2[DZ*L64];   // temp (IKC*P, KgR(ld L32), chol work, A*Pf)
  float Pf[DZ*L64];   // P_filt
  float C [DA*L64];   // mixed C
  float CP[DA*L64];   // C*P, then KgT after solve
  float Sm[DA*L32];   // innovation covariance
  float Bc[DZ*DU];    // mixed B
  float GI[3*HH], GH[3*HH];
  float h[HH], hnew[HH];
  float xbuf[DIN], ubuf[DU], abuf[DA];
  float alpha[KMIX];
  float pcol[DA];
  float zprev[DZ], z[DZ], zmean[DZ], zpred[DZ];
  float ahat[DA], rk[DA], afilt[DA], akhat[DA], apred[DA];
};

// ---- WMMA f32 16x16x4 tile GEMM ----
template<bool TA>
__device__ __forceinline__ v2f ld_a(const float* Am, int lda, int mrow, int ka) {
  v2f a;
  if (TA) { a.x = Am[(ka + 0) * lda + mrow]; a.y = Am[(ka + 1) * lda + mrow]; }
  else    { a.x = Am[mrow * lda + ka + 0];   a.y = Am[mrow * lda + ka + 1];   }
  return a;
}
template<bool TB>
__device__ __forceinline__ v2f ld_b(const float* Bm, int ldb, int ncol, int ka) {
  v2f b;
  if (TB) { b.x = Bm[ncol * ldb + ka + 0];   b.y = Bm[ncol * ldb + ka + 1];   }
  else    { b.x = Bm[(ka + 0) * ldb + ncol]; b.y = Bm[(ka + 1) * ldb + ncol]; }
  return b;
}

// 16x16 output tile; K-loop software-pipelined (load slice k+1 before WMMA on k)
template<bool TA, bool TB>
__device__ __forceinline__ void mm_tile16(const float* Am, int lda,
                                          const float* Bm, int ldb,
                                          float* Om, int ldo,
                                          int row0, int col0, int Kdim, int lane) {
  v8f acc = {};
  const int half = lane >> 4;
  const int l15  = lane & 15;
  const int mrow = row0 + l15;
  const int ncol = col0 + l15;
  const int kh   = half << 1;

  v2f a0 = ld_a<TA>(Am, lda, mrow, kh);
  v2f b0 = ld_b<TB>(Bm, ldb, ncol, kh);
  for (int k0 = 4; k0 < Kdim; k0 += 4) {
    v2f a1 = ld_a<TA>(Am, lda, mrow, k0 + kh);
    v2f b1 = ld_b<TB>(Bm, ldb, ncol, k0 + kh);
    acc = __builtin_amdgcn_wmma_f32_16x16x4_f32(false, a0, false, b0, (short)0, acc,
                                                false, false);
    a0 = a1; b0 = b1;
  }
  acc = __builtin_amdgcn_wmma_f32_16x16x4_f32(false, a0, false, b0, (short)0, acc,
                                              false, false);

  const int mb = row0 + (half << 3);
#pragma unroll
  for (int i = 0; i < 8; ++i) Om[(mb + i) * ldo + ncol] = acc[i];
}

// Full GEMM: Om(Mr x Nc) = op(A) * op(B), tiles distributed across 8 waves
template<bool TA, bool TB>
__device__ __forceinline__ void mm(const float* Am, int lda, const float* Bm, int ldb,
                                   float* Om, int ldo, int Mr, int Nc, int Kdim, int tid) {
  const int wave = tid >> 5, lane = tid & 31;
  const int tn = Nc >> 4;
  const int nt = (Mr >> 4) * tn;
  for (int t = wave; t < nt; t += NWAVE) {
    const int r0 = (t / tn) << 4;
    const int c0 = (t % tn) << 4;
    mm_tile16<TA, TB>(Am, lda, Bm, ldb, Om, ldo, r0, c0, Kdim, lane);
  }
}

__device__ __forceinline__ float sigmoidf_(float x) { return 1.0f / (1.0f + __expf(-x)); }

__device__ __forceinline__ void set_x(SMem* sm, const float* hob, int tid) {
  for (int i = tid; i < DIN; i += NTHR) {
    float v;
    if (i < DA)             v = sm->akhat[i];
    else if (i < DA + DOBS) v = hob[i - DA];
    else                    v = sm->ubuf[i - DA - DOBS];
    sm->xbuf[i] = v;
  }
  __syncthreads();
}

__device__ void alpha_net(SMem* sm, const float* W_ih, const float* W_hh,
                          const float* b_ih, const float* b_hh,
                          const float* W_out, const float* b_out, int tid) {
  for (int j = tid; j < 3 * HH; j += NTHR) {
    float gi = b_ih[j];
    const float* wi = W_ih + j * DIN;
    for (int c = 0; c < DIN; ++c) gi += wi[c] * sm->xbuf[c];
    float gh = b_hh[j];
    const float* wh = W_hh + j * HH;
    for (int c = 0; c < HH; ++c) gh += wh[c] * sm->h[c];
    sm->GI[j] = gi; sm->GH[j] = gh;
  }
  __syncthreads();
  for (int j = tid; j < HH; j += NTHR) {
    float r  = sigmoidf_(sm->GI[j] + sm->GH[j]);
    float zg = sigmoidf_(sm->GI[HH + j] + sm->GH[HH + j]);
    float n  = tanhf(sm->GI[2 * HH + j] + r * sm->GH[2 * HH + j]);
    sm->hnew[j] = (1.0f - zg) * n + zg * sm->h[j];
  }
  __syncthreads();
  for (int j = tid; j < HH; j += NTHR) sm->h[j] = sm->hnew[j];
  if (tid < KMIX) {
    float l = b_out[tid];
    const float* wo = W_out + tid * HH;
    for (int c = 0; c < HH; ++c) l += wo[c] * sm->hnew[c];
    sm->alpha[tid] = l;
  }
  __syncthreads();
  if (tid == 0) {
    float mx = sm->alpha[0];
    for (int k = 1; k < KMIX; ++k) mx = fmaxf(mx, sm->alpha[k]);
    float s = 0.f;
    for (int k = 0; k < KMIX; ++k) { float e = __expf(sm->alpha[k] - mx); sm->alpha[k] = e; s += e; }
    float inv = 1.0f / s;
    for (int k = 0; k < KMIX; ++k) sm->alpha[k] *= inv;
  }
  __syncthreads();
}

__device__ void mix(SMem* sm, const float* A_mats, const float* B_mats,
                    const float* C_mats, int tid) {
  float al[KMIX];
#pragma unroll
  for (int k = 0; k < KMIX; ++k) al[k] = sm->alpha[k];
  for (int i = tid; i < DZ * DZ; i += NTHR) {
    const int r = i / DZ, c = i % DZ;
    float v = 0.f;
#pragma unroll
    for (int k = 0; k < KMIX; ++k) v += al[k] * A_mats[k * DZ * DZ + i];
    sm->A[r * L64 + c] = v;
  }
  for (int i = tid; i < DZ * DU; i += NTHR) {
    float v = 0.f;
#pragma unroll
    for (int k = 0; k < KMIX; ++k) v += al[k] * B_mats[k * DZ * DU + i];
    sm->Bc[i] = v;
  }
  for (int i = tid; i < DA * DZ; i += NTHR) {
    const int r = i / DZ, c = i % DZ;
    float v = 0.f;
#pragma unroll
    for (int k = 0; k < KMIX; ++k) v += al[k] * C_mats[k * DA * DZ + i];
    sm->C[r * L64 + c] = v;
  }
  __syncthreads();
}

__device__ __forceinline__ void out_copy(float* dst, const float* src, int n, int tid) {
  for (int i = tid; i < n; i += NTHR) dst[i] = src[i];
}
// Gather from padded LDS matrix into dense global output
__device__ __forceinline__ void out_copy_mat(float* dst, const float* src,
                                             int rows, int cols, int ld, int tid) {
  for (int i = tid; i < rows * cols; i += NTHR)
    dst[i] = src[(i / cols) * ld + (i % cols)];
}

// Output layout offsets (flat concatenation, reference return order)
constexpr size_t BT = (size_t)BSZ * TLEN;
constexpr size_t O_ZFILT = 0;
constexpr size_t O_PFILT = O_ZFILT + BT * DZ;
constexpr size_t O_ZPRED = O_PFILT + BT * DZ * DZ;
constexpr size_t O_AFILT = O_ZPRED + BT * DZ;
constexpr size_t O_APRED = O_AFILT + BT * DA;
constexpr size_t O_PPRED = O_APRED + BT * DA;
constexpr size_t O_ALPHA = O_PPRED + BT * DZ * DZ;
constexpr size_t O_ASEQ  = O_ALPHA + BT * KMIX;
constexpr size_t O_BSEQ  = O_ASEQ  + BT * DZ * DZ;
constexpr size_t O_CSEQ  = O_BSEQ  + BT * DZ * DU;
constexpr size_t O_ZMEAN = O_CSEQ  + BT * DA * DZ;
constexpr size_t O_LTRIL = O_ZMEAN + BT * DZ;
constexpr size_t O_SPRED = O_LTRIL + BT * DZ * DZ;

__global__ void __launch_bounds__(NTHR, 1) kalman_gru_wmma_kernel(
    const float* __restrict__ a_seq, const float* __restrict__ h_obs,
    const float* __restrict__ u_seq, const float* __restrict__ mask,
    const float* __restrict__ A_mats, const float* __restrict__ B_mats,
    const float* __restrict__ C_mats, const float* __restrict__ a_0,
    const float* __restrict__ P_0,   const float* __restrict__ Qm,
    const float* __restrict__ Rm,    const float* __restrict__ W_ih,
    const float* __restrict__ W_hh,  const float* __restrict__ b_ih,
    const float* __restrict__ b_hh,  const float* __restrict__ W_out,
    const float* __restrict__ b_out, float* __restrict__ out) {
  extern __shared__ char smem_raw[];
  SMem* sm = (SMem*)smem_raw;
  const int tid = threadIdx.x;
  const int b = blockIdx.x;
  const float* hob = h_obs + (size_t)b * DOBS;

  // ---- init carry ----
  for (int i = tid; i < HH; i += NTHR) sm->h[i] = 0.f;
  for (int i = tid; i < DZ; i += NTHR) sm->zprev[i] = 0.f;
  for (int i = tid; i < DZ * DZ; i += NTHR)
    sm->P[(i / DZ) * L64 + (i % DZ)] = P_0[i];
  for (int i = tid; i < DU; i += NTHR) sm->ubuf[i] = u_seq[(size_t)b * TLEN * DU + i];
  for (int i = tid; i < DA; i += NTHR) sm->akhat[i] = a_0[i];
  __syncthreads();
  set_x(sm, hob, tid);
  alpha_net(sm, W_ih, W_hh, b_ih, b_hh, W_out, b_out, tid);
  mix(sm, A_mats, B_mats, C_mats, tid);

  for (int t = 0; t < TLEN; ++t) {
    const size_t bt = (size_t)b * TLEN + t;
    const float mval = mask[bt];
    for (int i = tid; i < DU; i += NTHR) sm->ubuf[i] = u_seq[bt * DU + i];
    for (int i = tid; i < DA; i += NTHR) sm->abuf[i] = a_seq[bt * DA + i];
    __syncthreads();

    // z = A z_prev + Bc u
    if (tid < DZ) {
      float v = 0.f;
      for (int c = 0; c < DZ; ++c) v += sm->A[tid * L64 + c] * sm->zprev[c];
      for (int c = 0; c < DU; ++c) v += sm->Bc[tid * DU + c] * sm->ubuf[c];
      sm->z[tid] = v;
    }
    __syncthreads();
    // a_hat = C z ; r = a - a_hat
    if (tid < DA) {
      float v = 0.f;
      for (int c = 0; c < DZ; ++c) v += sm->C[tid * L64 + c] * sm->z[c];
      sm->ahat[tid] = v;
      sm->rk[tid] = sm->abuf[tid] - v;
    }
    __syncthreads();

    // CP = C * P     (32x64, K=64)   [WMMA]
    mm<false, false>(sm->C, L64, sm->P, L64, sm->CP, L64, DA, DZ, DZ, tid);
    __syncthreads();
    // S = CP * C^T + R  (32x32, K=64) [WMMA]
    mm<false, true>(sm->CP, L64, sm->C, L64, sm->Sm, L32, DA, DA, DZ, tid);
    __syncthreads();
    for (int i = tid; i < DA * DA; i += NTHR)
      sm->Sm[(i / DA) * L32 + (i % DA)] += Rm[i];
    __syncthreads();
    out_copy_mat(out + O_SPRED + bt * DA * DA, sm->Sm, DA, DA, L32, tid);
    __syncthreads();

    // Gauss-Jordan: CP <- S^{-1} CP = Kg^T (32x64); Sm destroyed
    for (int p = 0; p < DA; ++p) {
      for (int r = tid; r < DA; r += NTHR) sm->pcol[r] = sm->Sm[r * L32 + p];
      __syncthreads();
      const float invp = 1.0f / sm->pcol[p];
      for (int c = tid; c < DA; c += NTHR) sm->Sm[p * L32 + c] *= invp;
      for (int c = tid; c < DZ; c += NTHR) sm->CP[p * L64 + c] *= invp;
      __syncthreads();
      for (int idx = tid; idx < DA * (DA + DZ); idx += NTHR) {
        const int r = idx / (DA + DZ);
        if (r == p) continue;
        const int c = idx % (DA + DZ);
        const float f = sm->pcol[r];
        if (c < DA) sm->Sm[r * L32 + c] -= f * sm->Sm[p * L32 + c];
        else        sm->CP[r * L64 + (c - DA)] -= f * sm->CP[p * L64 + (c - DA)];
      }
      __syncthreads();
    }
    // Kg *= mask
    for (int i = tid; i < DA * DZ; i += NTHR)
      sm->CP[(i / DZ) * L64 + (i % DZ)] *= mval;
    __syncthreads();

    // IKC: T1 = Kg * C (64x64, K=32; A read transposed from KgT) [WMMA], then I - T1
    mm<true, false>(sm->CP, L64, sm->C, L64, sm->T1, L64, DZ, DZ, DA, tid);
    __syncthreads();
    for (int i = tid; i < DZ * DZ; i += NTHR) {
      const int r = i / DZ, c = i % DZ;
      sm->T1[r * L64 + c] = ((r == c) ? 1.0f : 0.0f) - sm->T1[r * L64 + c];
    }
    __syncthreads();
    // T2 = IKC * P (64x64, K=64) [WMMA]
    mm<false, false>(sm->T1, L64, sm->P, L64, sm->T2, L64, DZ, DZ, DZ, tid);
    __syncthreads();
    // Pf = T2 * IKC^T (64x64, K=64) [WMMA]
    mm<false, true>(sm->T2, L64, sm->T1, L64, sm->Pf, L64, DZ, DZ, DZ, tid);
    __syncthreads();
    // T2(ld L32) = Kg * R (64x32, K=32; A transposed from KgT, B = R global) [WMMA]
    mm<true, false>(sm->CP, L64, Rm, DA, sm->T2, L32, DZ, DA, DA, tid);
    __syncthreads();
    // T1 = (Kg R) * Kg^T (64x64, K=32; B = KgT directly) [WMMA]
    mm<false, false>(sm->T2, L32, sm->CP, L64, sm->T1, L64, DZ, DZ, DA, tid);
    __syncthreads();
    // P_filt = sym(Pf + T1)  (each (i<=j) thread owns the symmetric pair)
    for (int idx = tid; idx < DZ * DZ; idx += NTHR) {
      const int i = idx / DZ, j = idx % DZ;
      if (i <= j) {
        const float v = 0.5f * (sm->Pf[i * L64 + j] + sm->T1[i * L64 + j] +
                                sm->Pf[j * L64 + i] + sm->T1[j * L64 + i]);
        sm->Pf[i * L64 + j] = v;
        sm->Pf[j * L64 + i] = v;
      }
    }
    __syncthreads();

    // z_mean = z + Kg r
    if (tid < DZ) {
      float v = sm->z[tid];
      for (int k = 0; k < DA; ++k) v += sm->CP[k * L64 + tid] * sm->rk[k];
      sm->zmean[tid] = v;
    }
    __syncthreads();
    // a_filt = C z_mean ; a_k_hat = m*a + (1-m)*a_filt
    if (tid < DA) {
      float v = 0.f;
      for (int c = 0; c < DZ; ++c) v += sm->C[tid * L64 + c] * sm->zmean[c];
      sm->afilt[tid] = v;
      sm->akhat[tid] = mval * sm->abuf[tid] + (1.0f - mval) * v;
    }
    __syncthreads();

    // Cholesky of Pf + 2*JIT*I into T2 (lower)
    for (int i = tid; i < DZ * DZ; i += NTHR) {
      const int r = i / DZ, c = i % DZ;
      sm->T2[r * L64 + c] = sm->Pf[r * L64 + c] + ((r == c) ? 2.0f * JIT : 0.0f);
    }
    __syncthreads();
    for (int j = 0; j < DZ; ++j) {
      if (tid == 0) sm->T2[j * L64 + j] = sqrtf(fmaxf(sm->T2[j * L64 + j], 1e-30f));
      __syncthreads();
      const float invd = 1.0f / sm->T2[j * L64 + j];
      for (int i = j + 1 + tid; i < DZ; i += NTHR) sm->T2[i * L64 + j] *= invd;
      __syncthreads();
      const int rem = DZ - j - 1;
      for (int idx = tid; idx < rem * rem; idx += NTHR) {
        const int r = j + 1 + idx / rem, c = j + 1 + idx % rem;
        if (c <= r) sm->T2[r * L64 + c] -= sm->T2[r * L64 + j] * sm->T2[c * L64 + j];
      }
      __syncthreads();
    }
    // write lower-tri L, z_filt/z_means, P_filt, a_filt, old C
    for (int i = tid; i < DZ * DZ; i += NTHR) {
      const int r = i / DZ, c = i % DZ;
      out[O_LTRIL + bt * DZ * DZ + i] = (c <= r) ? sm->T2[r * L64 + c] : 0.0f;
    }
    out_copy(out + O_ZFILT + bt * DZ, sm->zmean, DZ, tid);
    out_copy(out + O_ZMEAN + bt * DZ, sm->zmean, DZ, tid);
    out_copy_mat(out + O_PFILT + bt * DZ * DZ, sm->Pf, DZ, DZ, L64, tid);
    out_copy(out + O_AFILT + bt * DA, sm->afilt, DA, tid);
    out_copy_mat(out + O_CSEQ + bt * DA * DZ, sm->C, DA, DZ, L64, tid);
    __syncthreads();

    // alpha update with a_k_hat, then mix -> new A, Bc, C
    set_x(sm, hob, tid);
    alpha_net(sm, W_ih, W_hh, b_ih, b_hh, W_out, b_out, tid);
    mix(sm, A_mats, B_mats, C_mats, tid);
    out_copy(out + O_ALPHA + bt * KMIX, sm->alpha, KMIX, tid);
    out_copy_mat(out + O_ASEQ + bt * DZ * DZ, sm->A, DZ, DZ, L64, tid);
    out_copy(out + O_BSEQ + bt * DZ * DU, sm->Bc, DZ * DU, tid);

    // z_pred = A_n z_filt + B_n u ; a_pred = C_n z_pred
    if (tid < DZ) {
      float v = 0.f;
      for (int c = 0; c < DZ; ++c) v += sm->A[tid * L64 + c] * sm->zmean[c];
      for (int c = 0; c < DU; ++c) v += sm->Bc[tid * DU + c] * sm->ubuf[c];
      sm->zpred[tid] = v;
    }
    __syncthreads();
    if (tid < DA) {
      float v = 0.f;
      for (int c = 0; c < DZ; ++c) v += sm->C[tid * L64 + c] * sm->zpred[c];
      sm->apred[tid] = v;
    }
    out_copy(out + O_ZPRED + bt * DZ, sm->zpred, DZ, tid);
    __syncthreads();
    out_copy(out + O_APRED + bt * DA, sm->apred, DA, tid);

    // P_pred = sym(A_n Pf A_n^T + Q): T2 = A*Pf, P = T2*A^T [WMMA], then sym+Q
    mm<false, false>(sm->A, L64, sm->Pf, L64, sm->T2, L64, DZ, DZ, DZ, tid);
    __syncthreads();
    mm<false, true>(sm->T2, L64, sm->A, L64, sm->P, L64, DZ, DZ, DZ, tid);
    __syncthreads();
    for (int idx = tid; idx < DZ * DZ; idx += NTHR) {
      const int i = idx / DZ, j = idx % DZ;
      if (i <= j) {
        const float v = 0.5f * (sm->P[i * L64 + j] + sm->P[j * L64 + i]) +
                        0.5f * (Qm[i * DZ + j] + Qm[j * DZ + i]);
        sm->P[i * L64 + j] = v;
        sm->P[j * L64 + i] = v;
      }
    }
    __syncthreads();
    out_copy_mat(out + O_PPRED + bt * DZ * DZ, sm->P, DZ, DZ, L64, tid);

    // carry: z_prev <- z_filt
    for (int i = tid; i < DZ; i += NTHR) sm->zprev[i] = sm->zmean[i];
    __syncthreads();
  }
}

extern "C" void kernel_launch(void* const* d_in, const int* in_sizes, int n_in,
                              void* d_out, int out_size, void* d_ws, size_t ws_size,
                              hipStream_t stream) {
  const float* a_seq  = (const float*)d_in[0];
  const float* h_obs  = (const float*)d_in[1];
  const float* u_seq  = (const float*)d_in[2];
  const float* maskp  = (const float*)d_in[3];
  const float* A_mats = (const float*)d_in[4];
  const float* B_mats = (const float*)d_in[5];
  const float* C_mats = (const float*)d_in[6];
  const float* a_0    = (const float*)d_in[7];
  const float* P_0    = (const float*)d_in[8];
  const float* Qm     = (const float*)d_in[9];
  const float* Rm     = (const float*)d_in[10];
  const float* W_ih   = (const float*)d_in[11];
  const float* W_hh   = (const float*)d_in[12];
  const float* b_ih   = (const float*)d_in[13];
  const float* b_hh   = (const float*)d_in[14];
  const float* W_out  = (const float*)d_in[15];
  const float* b_out  = (const float*)d_in[16];
  float* out = (float*)d_out;

  kalman_gru_wmma_kernel<<<dim3(BSZ), dim3(NTHR), sizeof(SMem), stream>>>(
      a_seq, h_obs, u_seq, maskp, A_mats, B_mats, C_mats, a_0, P_0, Qm, Rm,
      W_ih, W_hh, b_ih, b_hh, W_out, b_out, out);
}